// FrequencyShiftLayer_61701500175244
// MI455X (gfx1250) — compile-verified
//
#include <hip/hip_runtime.h>

// y = exp(i*LZ*DPHI) * x  (ifft2(fft2(x)*c) == c*x by linearity)
// theta = 0.5 * 0.1 = 0.05
#define COS_T 0.99875026039496624654f
#define SIN_T 0.04997916927067833474f

typedef float v4f __attribute__((ext_vector_type(4)));

__global__ __launch_bounds__(256) void freqshift_rot4(
    const float* __restrict__ xr,
    const float* __restrict__ xi,
    float* __restrict__ yr,
    float* __restrict__ yi,
    long long n4)
{
    long long i = (long long)blockIdx.x * blockDim.x + threadIdx.x;
    if (i >= n4) return;

    const v4f* xr4 = (const v4f*)xr;
    const v4f* xi4 = (const v4f*)xi;

    // Non-temporal: stream (512 MB total) exceeds 192 MB L2, nothing is reused.
    v4f a = __builtin_nontemporal_load(xr4 + i);  // global_load_b128 th:NT
    v4f b = __builtin_nontemporal_load(xi4 + i);

    v4f re = a * COS_T - b * SIN_T;
    v4f im = a * SIN_T + b * COS_T;

    __builtin_nontemporal_store(re, (v4f*)yr + i); // global_store_b128 th:NT
    __builtin_nontemporal_store(im, (v4f*)yi + i);
}

// Scalar tail (unused for B*M*N = 32*1024*1024, kept for robustness).
__global__ __launch_bounds__(64) void freqshift_rot_tail(
    const float* __restrict__ xr,
    const float* __restrict__ xi,
    float* __restrict__ yr,
    float* __restrict__ yi,
    long long start, long long n)
{
    long long i = start + (long long)blockIdx.x * blockDim.x + threadIdx.x;
    if (i >= n) return;
    float a = xr[i];
    float b = xi[i];
    yr[i] = a * COS_T - b * SIN_T;
    yi[i] = a * SIN_T + b * COS_T;
}

extern "C" void kernel_launch(void* const* d_in, const int* in_sizes, int n_in,
                              void* d_out, int out_size, void* d_ws, size_t ws_size,
                              hipStream_t stream) {
    (void)n_in; (void)out_size; (void)d_ws; (void)ws_size;

    const float* xr = (const float*)d_in[0];
    const float* xi = (const float*)d_in[1];

    long long n = (long long)in_sizes[0];       // B*M*N = 33,554,432
    float* yr = (float*)d_out;                  // [0]  real plane
    float* yi = (float*)d_out + n;              // [1]  imag plane

    long long n4 = n >> 2;                      // float4 count
    const int block = 256;                      // 8 wave32 waves per block
    long long grid = (n4 + block - 1) / block;
    if (grid > 0) {
        freqshift_rot4<<<(unsigned)grid, block, 0, stream>>>(xr, xi, yr, yi, n4);
    }

    long long done = n4 << 2;
    long long rem = n - done;
    if (rem > 0) {
        long long tgrid = (rem + 63) / 64;
        freqshift_rot_tail<<<(unsigned)tgrid, 64, 0, stream>>>(xr, xi, yr, yi, done, n);
    }
}